// Avg_Pooling_Variable_5712306503683
// MI455X (gfx1250) — compile-verified
//
#include <hip/hip_runtime.h>
#include <stdint.h>

// Masked mean pooling, MI455X (gfx1250).
// Bandwidth-bound (0.5 FLOP/byte): design = async global->LDS double-buffered
// staging (CDNA5 path) + WMMA f32 16x16x4 ones-vector reduction.

typedef float v2f __attribute__((ext_vector_type(2)));
typedef float v8f __attribute__((ext_vector_type(8)));

#ifndef __has_builtin
#define __has_builtin(x) 0
#endif

#if __has_builtin(__builtin_amdgcn_wmma_f32_16x16x4_f32)
#define HAS_WMMA 1
#else
#define HAS_WMMA 0
#endif

constexpr int Bn = 32, Sn = 2048, Dn = 1024;
constexpr int DTILE = 128;       // d-columns per workgroup (8 waves x 16)
constexpr int ROWS = 32;         // s-rows per staged block (16 KB payload)
constexpr int LSTRIDE = 144;     // LDS floats/row; 144%64==16 -> half-wave bank-disjoint
constexpr int SCHUNKS = 8;
constexpr int CHUNK = Sn / SCHUNKS;  // 256

__device__ __forceinline__ void wait_async0() {
#if __has_builtin(__builtin_amdgcn_s_wait_asynccnt)
  __builtin_amdgcn_s_wait_asynccnt(0);
#else
  asm volatile("s_wait_asynccnt 0x0" ::: "memory");
#endif
}

// Stage one block: rows [0, valid) async-copied global->LDS, rows [valid, ROWS)
// zero-filled (this implements the length mask; WMMA then needs no masking).
// Wave w owns rows {w, w+8, w+16, w+24}; each async moves 512B (16B/lane).
__device__ __forceinline__ void stage_block(const float* __restrict__ grow0,
                                            float* buf, int valid,
                                            int wave, int lane) {
#pragma unroll
  for (int i = 0; i < 4; ++i) {
    const int r = wave + i * 8;
    float* lrow = buf + r * LSTRIDE;
    if (r < valid) {
      const float* g = grow0 + (size_t)r * Dn + lane * 4;
      const uint32_t loff =                 // low 32 bits of a generic LDS
          (uint32_t)(uintptr_t)lrow +       // pointer == LDS byte address
          (uint32_t)(lane * 16);
      asm volatile("global_load_async_to_lds_b128 %0, %1, off"
                   :: "v"(loff), "v"((uint64_t)(uintptr_t)g)
                   : "memory");
    } else {
      ((float4*)lrow)[lane] = make_float4(0.f, 0.f, 0.f, 0.f);  // ds_store_b128
    }
  }
}

__global__ __launch_bounds__(256) void pool_pass1(
    const float* __restrict__ feat, const int* __restrict__ lengths,
    float* __restrict__ dst, int chunk_rows, int finalize) {
  __shared__ float buf[2 * ROWS * LSTRIDE];  // 36 KB, double buffer

  const int dtile = blockIdx.x;             // 0..7
  const int b     = blockIdx.y;             // 0..31
  const int chunk = blockIdx.z;
  const int tid   = threadIdx.x;
  const int wave  = tid >> 5;
  const int lane  = tid & 31;

  const int len = lengths[b];
  const int s0c = chunk * chunk_rows;
  int valid = len - s0c;
  if (valid > chunk_rows) valid = chunk_rows;
  const int nblk = (valid > 0) ? (valid + ROWS - 1) / ROWS : 0;  // uniform in WG

  const float* gtile = feat + (size_t)b * Sn * Dn + (size_t)dtile * DTILE;

  v8f acc = {0.f, 0.f, 0.f, 0.f, 0.f, 0.f, 0.f, 0.f};
  float facc = 0.f;
  (void)facc; (void)acc;

  if (nblk > 0) {
    int v0 = valid > ROWS ? ROWS : valid;
    stage_block(gtile + (size_t)s0c * Dn, buf, v0, wave, lane);

    const int col  = lane & 15;
    const int half = lane >> 4;

    for (int blk = 0; blk < nblk; ++blk) {
      const float* cur = buf + (blk & 1) * (ROWS * LSTRIDE);

      wait_async0();        // my block-blk rows landed in LDS
      __syncthreads();      // everyone's landed; prior consumers of buf_next done

      if (blk + 1 < nblk) {  // prefetch next block (overlaps with consume below)
        int vr = valid - (blk + 1) * ROWS;
        if (vr > ROWS) vr = ROWS;
        stage_block(gtile + (size_t)(s0c + (blk + 1) * ROWS) * Dn,
                    buf + ((blk + 1) & 1) * (ROWS * LSTRIDE), vr, wave, lane);
      }

      const float* bcol = cur + wave * 16 + col;
#if HAS_WMMA
      // D = ones(16x4) x B(4x16) + C : every row of D = column sums of B.
      // Row-permutation invariant, so the B K-slot mapping below (any
      // bijection of 4 rows onto {vgpr,half}) is safe; masking is via the
      // zero-filled LDS rows. Bank-disjoint halves thanks to LSTRIDE=144.
#pragma unroll
      for (int r = 0; r < ROWS; r += 4) {
        v2f av = {1.0f, 1.0f};
        v2f bv = {bcol[(r + half) * LSTRIDE],
                  bcol[(r + 2 + half) * LSTRIDE]};
        acc = __builtin_amdgcn_wmma_f32_16x16x4_f32(
            false, av, false, bv, (short)0, acc, false, false);
      }
#else
#pragma unroll
      for (int r = 0; r < ROWS; ++r) facc += bcol[r * LSTRIDE];
#endif
      __syncthreads();  // consumers done before anyone restages this buffer
    }
  }

#if HAS_WMMA
  const float result = acc[0];  // C/D layout: VGPR0, lanes 0-15 = row M=0
#else
  const float result = facc;
#endif
  if (lane < 16) {
    const size_t o = (size_t)b * Dn + dtile * DTILE + wave * 16 + (lane & 15);
    if (finalize) {
      dst[o] = result / (float)len;
    } else {
      dst[(size_t)chunk * (Bn * Dn) + o] = result;
    }
  }
}

__global__ __launch_bounds__(256) void pool_pass2(
    const float* __restrict__ part, const int* __restrict__ lengths,
    float* __restrict__ out) {
  const int idx = blockIdx.x * 256 + threadIdx.x;  // 0..32767
  const int b = idx >> 10;
  float s = 0.f;
#pragma unroll
  for (int c = 0; c < SCHUNKS; ++c) s += part[(size_t)c * (Bn * Dn) + idx];
  out[idx] = s / (float)lengths[b];
}

extern "C" void kernel_launch(void* const* d_in, const int* in_sizes, int n_in,
                              void* d_out, int out_size, void* d_ws,
                              size_t ws_size, hipStream_t stream) {
  const float* feat = (const float*)d_in[0];
  const int* lens   = (const int*)d_in[1];
  float* out        = (float*)d_out;

  const size_t need = (size_t)SCHUNKS * Bn * Dn * sizeof(float);  // 1 MiB
  if (ws_size >= need) {
    float* part = (float*)d_ws;
    dim3 grid(Dn / DTILE, Bn, SCHUNKS);  // 8 x 32 x 8 = 2048 workgroups
    pool_pass1<<<grid, 256, 0, stream>>>(feat, lens, part, CHUNK, 0);
    pool_pass2<<<(Bn * Dn) / 256, 256, 0, stream>>>(part, lens, out);
  } else {
    dim3 grid(Dn / DTILE, Bn, 1);
    pool_pass1<<<grid, 256, 0, stream>>>(feat, lens, out, Sn, 1);
  }
}